// Encoder_72851235275124
// MI455X (gfx1250) — compile-verified
//
#include <hip/hip_runtime.h>
#include <hip/hip_bf16.h>

typedef __attribute__((ext_vector_type(2))) float v2f;
typedef __attribute__((ext_vector_type(8))) float v8f;

#define BN_EPS 1e-5f

__device__ __forceinline__ float bn_relu(float acc, int oc, const float* bia,
                                         const float* gam, const float* bet,
                                         const float* mu, const float* var)
{
    float sc = gam[oc] * rsqrtf(var[oc] + BN_EPS);
    return fmaxf((acc + bia[oc] - mu[oc]) * sc + bet[oc], 0.f);
}

// ---------------------------------------------------------------------------
// Fused conv1d(k=3,pad=1)+BN+ReLU via V_WMMA_F32_16X16X4_F32.  Cin % 4 == 0.
// K-loop over channel groups of 4 (12 taps = 3 WMMA); per-lane tap offsets
// precomputed -> branch-free, division-free inner loop.
// WG: 128 thr = 4 waves; tile 16 out-ch x 64 positions. grid=(L/64,Cout/16,B)
// ---------------------------------------------------------------------------
__global__ __launch_bounds__(128)
void conv3_bn_relu_wmma(const float* __restrict__ x, const float* __restrict__ w,
                        const float* __restrict__ bia, const float* __restrict__ gam,
                        const float* __restrict__ bet, const float* __restrict__ mu,
                        const float* __restrict__ var, float* __restrict__ y,
                        int Cin, int Cout, int L)
{
    __shared__ float xs[128 * 66];            // [Cin][66] input tile incl. halo
    const int bb  = blockIdx.z;
    const int oc0 = blockIdx.y << 4;
    const int p0  = blockIdx.x << 6;
    const int tid = threadIdx.x;
    const int K   = Cin * 3;

    for (int i = tid; i < Cin * 66; i += 128) {
        int ci = i / 66, t = i - ci * 66;
        int pos = p0 - 1 + t;
        xs[i] = (pos >= 0 && pos < L) ? x[((long)bb * Cin + ci) * L + pos] : 0.f;
    }
    __syncthreads();

    const int wv    = tid >> 5;
    const int lane  = tid & 31;
    const int mrow  = lane & 15;             // A row (out channel)
    const int kh    = (lane >> 4) << 1;      // lane-half K offset: 0 or 2
    const int tbase = (wv << 4) + (lane & 15);  // position within LDS tile

    // Per-lane tap decomposition of the 12 k-indices in a 4-channel group:
    // for WMMA s (0..2), element j (0..1): kk = 4s + kh + j -> (ci, kw).
    int ciO[3][2], kwO[3][2];
#pragma unroll
    for (int s = 0; s < 3; ++s)
#pragma unroll
        for (int j = 0; j < 2; ++j) {
            int kk = 4 * s + kh + j;
            ciO[s][j] = kk / 3;
            kwO[s][j] = kk - 3 * ciO[s][j];
        }

    const float* wrow = w + (long)(oc0 + mrow) * K;  // rows contiguous in K=I*3
    v8f c = {};
    const int ngrp = Cin >> 2;
    for (int g = 0; g < ngrp; ++g) {
        const int ci0   = g << 2;
        const int kbase = g * 12;
#pragma unroll
        for (int s = 0; s < 3; ++s) {
            v2f a, b;
            a.x = wrow[kbase + 4 * s + kh];
            a.y = wrow[kbase + 4 * s + kh + 1];
            b.x = xs[(ci0 + ciO[s][0]) * 66 + tbase + kwO[s][0]];
            b.y = xs[(ci0 + ciO[s][1]) * 66 + tbase + kwO[s][1]];
            c = __builtin_amdgcn_wmma_f32_16x16x4_f32(false, a, false, b, (short)0,
                                                      c, false, false);
        }
    }

    const int pos = p0 + tbase;
#pragma unroll
    for (int j = 0; j < 8; ++j) {
        int oc = oc0 + j + ((lane >> 4) << 3);
        y[((long)bb * Cout + oc) * L + pos] = bn_relu(c[j], oc, bia, gam, bet, mu, var);
    }
}

// ---------------------------------------------------------------------------
// Specialized first conv: Cin = 1 (K = 3).  One WMMA; kh in {0,2} so the K=3
// boundary reduces to a cndmask (a.y/b.y zeroed when kh==2). grid=(L/64,2,B)
// ---------------------------------------------------------------------------
__global__ __launch_bounds__(128)
void conv3_c1_bn_relu_wmma(const float* __restrict__ x, const float* __restrict__ w,
                           const float* __restrict__ bia, const float* __restrict__ gam,
                           const float* __restrict__ bet, const float* __restrict__ mu,
                           const float* __restrict__ var, float* __restrict__ y,
                           int Cout, int L)
{
    __shared__ float xs[66];
    const int bb  = blockIdx.z;
    const int oc0 = blockIdx.y << 4;
    const int p0  = blockIdx.x << 6;
    const int tid = threadIdx.x;

    if (tid < 66) {
        int pos = p0 - 1 + tid;
        xs[tid] = (pos >= 0 && pos < L) ? x[(long)bb * L + pos] : 0.f;
    }
    __syncthreads();

    const int wv    = tid >> 5;
    const int lane  = tid & 31;
    const int mrow  = lane & 15;
    const int kh    = (lane >> 4) << 1;
    const int tbase = (wv << 4) + (lane & 15);

    const float* wrow = w + (long)(oc0 + mrow) * 3;
    v2f a, b;
    a.x = wrow[kh];
    a.y = (kh == 0) ? wrow[1] : 0.f;
    b.x = xs[tbase + kh];
    b.y = (kh == 0) ? xs[tbase + 1] : 0.f;
    v8f c = {};
    c = __builtin_amdgcn_wmma_f32_16x16x4_f32(false, a, false, b, (short)0, c,
                                              false, false);

    const int pos = p0 + tbase;
#pragma unroll
    for (int j = 0; j < 8; ++j) {
        int oc = oc0 + j + ((lane >> 4) << 3);
        y[((long)bb * Cout + oc) * L + pos] = bn_relu(c[j], oc, bia, gam, bet, mu, var);
    }
}

// ---------------------------------------------------------------------------
// 1x1 latent conv: Cout=64, bias only.  K = Cin (32/64/128, %4==0).
// grid = (L/64, 4, B), 128 threads.
// ---------------------------------------------------------------------------
__global__ __launch_bounds__(128)
void conv1_lat_wmma(const float* __restrict__ x, const float* __restrict__ w,
                    const float* __restrict__ bia, float* __restrict__ y,
                    int Cin, int L)
{
    const int bb  = blockIdx.z;
    const int oc0 = blockIdx.y << 4;
    const int p0  = blockIdx.x << 6;
    const int tid = threadIdx.x;
    const int wv   = tid >> 5;
    const int lane = tid & 31;
    const int mrow = lane & 15;
    const int ncol = lane & 15;
    const int kh   = (lane >> 4) << 1;
    const int pw   = p0 + (wv << 4);

    const float* wrow = w + (long)(oc0 + mrow) * Cin;   // [64, Cin, 1]
    const float* xcol = x + (long)bb * Cin * L + pw + ncol;
    v8f c = {};
#pragma unroll 8
    for (int k0 = 0; k0 < Cin; k0 += 4) {
        v2f a, b;
        a.x = wrow[k0 + kh];
        a.y = wrow[k0 + kh + 1];
        b.x = xcol[(long)(k0 + kh) * L];
        b.y = xcol[(long)(k0 + kh + 1) * L];
        c = __builtin_amdgcn_wmma_f32_16x16x4_f32(false, a, false, b, (short)0, c,
                                                  false, false);
    }
#pragma unroll
    for (int j = 0; j < 8; ++j) {
        int oc = oc0 + j + ((lane >> 4) << 3);
        y[((long)bb * 64 + oc) * L + pw + ncol] = c[j] + bia[oc];
    }
}

// ---------------------------------------------------------------------------
// Codebook squared norms (512 x 64), once.
// ---------------------------------------------------------------------------
__global__ void cnorm_kernel(const float* __restrict__ cb, float* __restrict__ cn)
{
    int n = blockIdx.x * blockDim.x + threadIdx.x;
    if (n < 512) {
        float s = 0.f;
        const float* r = cb + n * 64;
#pragma unroll 8
        for (int c = 0; c < 64; ++c) s += r[c] * r[c];
        cn[n] = s;
    }
}

// ---------------------------------------------------------------------------
// VQ: WG = 256 thr = 8 waves, 16 positions vs all 512 codes.  A (xf tile) is
// hoisted to registers once per wave (identical for all 4 N-tiles); K=64 loop
// fully unrolled.  Argmin via order-preserving u64 key + ds_min_u64.
// grid = (L/16, B)
// ---------------------------------------------------------------------------
__global__ __launch_bounds__(256)
void vq_wmma(const float* __restrict__ xlat, const float* __restrict__ cb,
             const float* __restrict__ cn, float* __restrict__ zq,
             float* __restrict__ z, float* __restrict__ idxout, int L)
{
    __shared__ float xs[16 * 65];
    __shared__ unsigned long long keymin[16];

    const int bb   = blockIdx.y;
    const int pos0 = blockIdx.x << 4;
    const int tid  = threadIdx.x;

    for (int i = tid; i < 1024; i += 256) {
        int m = i & 15, ch = i >> 4;
        xs[m * 65 + ch] = xlat[((long)bb * 64 + ch) * L + pos0 + m];
    }
    if (tid < 16) keymin[tid] = ~0ULL;
    __syncthreads();

    const int wv   = tid >> 5;
    const int lane = tid & 31;
    const int mrow = lane & 15;
    const int kh   = (lane >> 4) << 1;

    // A-matrix fragment: constant across all N-tiles -> load once.
    v2f areg[16];
#pragma unroll
    for (int q = 0; q < 16; ++q) {
        areg[q].x = xs[mrow * 65 + 4 * q + kh];
        areg[q].y = xs[mrow * 65 + 4 * q + kh + 1];
    }

#pragma unroll
    for (int t = 0; t < 4; ++t) {
        const int n0 = ((wv << 2) + t) << 4;
        const int n  = n0 + (lane & 15);
        const float* crow = cb + (long)n * 64;
        v8f acc = {};
#pragma unroll
        for (int q = 0; q < 16; ++q) {
            v2f b;
            b.x = crow[4 * q + kh];
            b.y = crow[4 * q + kh + 1];
            acc = __builtin_amdgcn_wmma_f32_16x16x4_f32(false, areg[q], false, b,
                                                        (short)0, acc, false, false);
        }
        const float cnn = cn[n];
#pragma unroll
        for (int j = 0; j < 8; ++j) {
            int m = j + ((lane >> 4) << 3);
            float d = cnn - 2.f * acc[j];     // ||x||^2 row-constant: dropped
            unsigned u = __float_as_uint(d);
            u = (u & 0x80000000u) ? ~u : (u | 0x80000000u);
            unsigned long long key = ((unsigned long long)u << 32) | (unsigned)n;
            atomicMin(&keymin[m], key);       // ds_min_u64
        }
    }
    __syncthreads();

    for (int i = tid; i < 1024; i += 256) {
        int m = i & 15, ch = i >> 4;
        unsigned n = (unsigned)(keymin[m] & 0xFFFFFFFFULL);
        float xf = xs[m * 65 + ch];
        float zv = cb[(size_t)n * 64 + ch];
        long off = ((long)bb * 64 + ch) * L + pos0 + m;
        z[off]  = zv;
        zq[off] = xf + (zv - xf);
    }
    if (tid < 16)
        idxout[(long)bb * L + pos0 + tid] =
            (float)(unsigned)(keymin[tid] & 0xFFFFFFFFULL);
}

// ---------------------------------------------------------------------------
// maxpool1d rate 4 over contiguous last dim.
// ---------------------------------------------------------------------------
__global__ void maxpool4_kernel(const float* __restrict__ x, float* __restrict__ y, int n)
{
    for (int i = blockIdx.x * blockDim.x + threadIdx.x; i < n;
         i += gridDim.x * blockDim.x) {
        const float* p = x + (long)i * 4;
        y[i] = fmaxf(fmaxf(p[0], p[1]), fmaxf(p[2], p[3]));
    }
}

// ---------------------------------------------------------------------------
// Inputs: signal | 4x(w1,b1,g1,be1,w2,b2,g2,be2) | 4x(lw,lb) | codebook |
// 4x(m1,v1,m2,v2).  Output: zq3..zq0 | z3..z0 | x3..x0 | idx3..idx0.
// ---------------------------------------------------------------------------
extern "C" void kernel_launch(void* const* d_in, const int* in_sizes, int n_in,
                              void* d_out, int out_size, void* d_ws, size_t ws_size,
                              hipStream_t stream)
{
    (void)in_sizes; (void)n_in; (void)out_size; (void)ws_size;
    const float* signal   = (const float*)d_in[0];
    const float* codebook = (const float*)d_in[41];
    float* out = (float*)d_out;

    float* bufA = (float*)d_ws;                 // <= 8,388,608 floats
    float* bufB = bufA + 8388608;
    float* bufC = bufB + 8388608;               // pooled, <= 2,097,152 floats
    float* cn   = bufC + 2097152;               // 512 floats

    const int Ls[4] = {16384, 4096, 1024, 256};
    const int Fs[4] = {32, 64, 128, 128};

    long e[4];
    for (int i = 0; i < 4; ++i) e[i] = 1024L * Ls[i];
    const long S = e[0] + e[1] + e[2] + e[3];
    long zqoff[4], idxoff[4];
    zqoff[3] = 0;              idxoff[3] = 0;
    zqoff[2] = e[3];           idxoff[2] = 16L * Ls[3];
    zqoff[1] = e[3] + e[2];    idxoff[1] = 16L * (Ls[3] + Ls[2]);
    zqoff[0] = e[3] + e[2] + e[1];
    idxoff[0] = 16L * (Ls[3] + Ls[2] + Ls[1]);

    cnorm_kernel<<<2, 256, 0, stream>>>(codebook, cn);

    const float* cur = signal;
    int Cin = 1;
    for (int i = 0; i < 4; ++i) {
        const int F = Fs[i], L = Ls[i];
        const float* w1  = (const float*)d_in[1 + 8 * i + 0];
        const float* b1  = (const float*)d_in[1 + 8 * i + 1];
        const float* g1  = (const float*)d_in[1 + 8 * i + 2];
        const float* be1 = (const float*)d_in[1 + 8 * i + 3];
        const float* w2  = (const float*)d_in[1 + 8 * i + 4];
        const float* b2  = (const float*)d_in[1 + 8 * i + 5];
        const float* g2  = (const float*)d_in[1 + 8 * i + 6];
        const float* be2 = (const float*)d_in[1 + 8 * i + 7];
        const float* m1  = (const float*)d_in[42 + 4 * i + 0];
        const float* v1  = (const float*)d_in[42 + 4 * i + 1];
        const float* m2  = (const float*)d_in[42 + 4 * i + 2];
        const float* v2  = (const float*)d_in[42 + 4 * i + 3];
        const float* lw  = (const float*)d_in[33 + 2 * i + 0];
        const float* lb  = (const float*)d_in[33 + 2 * i + 1];

        dim3 gc(L / 64, F / 16, 16);
        if (i == 0)
            conv3_c1_bn_relu_wmma<<<gc, 128, 0, stream>>>(cur, w1, b1, g1, be1,
                                                          m1, v1, bufA, F, L);
        else
            conv3_bn_relu_wmma<<<gc, 128, 0, stream>>>(cur, w1, b1, g1, be1, m1, v1,
                                                       bufA, Cin, F, L);
        conv3_bn_relu_wmma<<<gc, 128, 0, stream>>>(bufA, w2, b2, g2, be2, m2, v2,
                                                   bufB, F, F, L);

        float* xsec = out + 2 * S + zqoff[i];
        conv1_lat_wmma<<<dim3(L / 64, 4, 16), 128, 0, stream>>>(bufB, lw, lb, xsec,
                                                                F, L);
        vq_wmma<<<dim3(L / 16, 16), 256, 0, stream>>>(xsec, codebook, cn,
                                                      out + zqoff[i],
                                                      out + S + zqoff[i],
                                                      out + 3 * S + idxoff[i], L);
        if (i < 3) {
            int n = 16 * F * (L / 4);
            maxpool4_kernel<<<(n + 255) / 256, 256, 0, stream>>>(bufB, bufC, n);
            cur = bufC;
            Cin = F;
        }
    }
}